// ILRMA_13340168421407
// MI455X (gfx1250) — compile-verified
//
#include <hip/hip_runtime.h>
#include <math.h>

#define MCH 2
#define JF 2000
#define IB 2049
#define KB 8
#define NITER 5
#define NMF_EPS 1e-20f
#define IP_EPS  1e-20f

typedef __attribute__((ext_vector_type(2))) float v2f;
typedef __attribute__((ext_vector_type(8))) float v8f;

// Fast reciprocal: v_rcp_f32 + 1 Newton-Raphson step (~1 ulp; reference divides in f32)
__device__ __forceinline__ float fastrcp(float x) {
  float r = __builtin_amdgcn_rcpf(x);
  return r * (2.0f - x * r);
}

// ---------------------------------------------------------------------------
// X (M,J,I,2) -> planar Xre/Xim (I,M,J), LDS tile transpose (coalesced both ways)
__global__ void k_prep_x(const float* __restrict__ X,
                         float* __restrict__ Xre, float* __restrict__ Xim) {
  __shared__ float sre[32][33];
  __shared__ float sim[32][33];
  int m  = blockIdx.z;
  int i0 = blockIdx.y * 32;
  int j0 = blockIdx.x * 32;
  int tx = threadIdx.x;   // 0..31
  int ty = threadIdx.y;   // 0..7
  for (int s = 0; s < 4; ++s) {
    int jl = ty + 8 * s;
    int i = i0 + tx, j = j0 + jl;
    float re = 0.f, im = 0.f;
    if (i < IB && j < JF) {
      const float* p = X + (((size_t)m * JF + j) * IB + i) * 2;
      re = p[0]; im = p[1];
    }
    sre[jl][tx] = re; sim[jl][tx] = im;
  }
  __syncthreads();
  for (int s = 0; s < 4; ++s) {
    int il = ty + 8 * s;
    int i = i0 + il, j = j0 + tx;
    if (i < IB && j < JF) {
      size_t o = ((size_t)i * MCH + m) * JF + j;
      Xre[o] = sre[tx][il];
      Xim[o] = sim[tx][il];
    }
  }
}

// W (I,2,2) complex -> identity
__global__ void k_init_w(float* __restrict__ W) {
  int i = blockIdx.x * 256 + threadIdx.x;
  if (i >= IB) return;
  float* wp = W + (size_t)i * 8;
  wp[0] = 1.f; wp[1] = 0.f; wp[2] = 0.f; wp[3] = 0.f;
  wp[4] = 0.f; wp[5] = 0.f; wp[6] = 1.f; wp[7] = 0.f;
}

// P[i,j] = |W[i,n,0]*x0 + W[i,n,1]*x1|^2
__global__ void k_compute_p(const float* __restrict__ Xre, const float* __restrict__ Xim,
                            const float* __restrict__ W, float* __restrict__ P, int n) {
  int j = blockIdx.x * 256 + threadIdx.x;
  int i = blockIdx.y;
  if (j >= JF) return;
  const float* w = W + (size_t)i * 8 + (size_t)n * 4;
  float w0r = w[0], w0i = w[1], w1r = w[2], w1i = w[3];
  size_t o0 = ((size_t)i * 2 + 0) * JF + j;
  size_t o1 = ((size_t)i * 2 + 1) * JF + j;
  float a = Xre[o0], b = Xim[o0], c = Xre[o1], d = Xim[o1];
  float yr = w0r * a - w0i * b + w1r * c - w1i * d;
  float yi = w0r * b + w0i * a + w1r * d + w1i * c;
  P[(size_t)i * JF + j] = yr * yr + yi * yi;
}

// Vpad[16][J]: rows 0..7 = V[:, :, n], rows 8..15 = 0 (for 16-wide WMMA B)
__global__ void k_pack_vpad(const float* __restrict__ V, float* __restrict__ Vpad, int n) {
  int idx = blockIdx.x * 256 + threadIdx.x;
  if (idx >= 16 * JF) return;
  int k = idx / JF, j = idx - k * JF;
  Vpad[idx] = (k < KB) ? V[((size_t)k * JF + j) * MCH + n] : 0.f;
}

// ---------------------------------------------------------------------------
// WMMA T-update:  numT = (P/R^2) @ V^T ,  denT = (1/R) @ V^T  as
// D[16 bins x 16 cols] += A[16x4] x B[4x16], V_WMMA_F32_16X16X4_F32.
// Two independent accumulator chains (even/odd 4-col chunks) for WMMA ILP.
__global__ void k_t_update(const float* __restrict__ P, const float* __restrict__ Vpad,
                           float* __restrict__ T, int n) {
  int lane = threadIdx.x & 31;
  int warp = threadIdx.x >> 5;
  int tile = blockIdx.x * 4 + warp;
  int base = tile * 16;
  if (base >= IB) return;               // uniform per wave
  int row = lane & 15;                  // A row / B-D column
  int hi  = lane >> 4;                  // half-wave select
  int bin = base + row;
  int binc = bin < IB ? bin : IB - 1;   // clamp loads (rows independent in D)
  float tk[KB];
#pragma unroll
  for (int k = 0; k < KB; ++k) tk[k] = T[((size_t)binc * KB + k) * MCH + n];
  const float* prow = P + (size_t)binc * JF;
  const float* brow = Vpad + (size_t)row * JF;  // B column = row var
  v8f dnum0 = {}, dden0 = {}, dnum1 = {}, dden1 = {};
  int jo = 2 * hi;                      // A: lane holds K-cols {2hi,2hi+1}; B symmetric
  for (int j = 0; j < JF; j += 8) {
    int ja = j + jo;                    // chunk 0
    int jb = ja + 4;                    // chunk 1
    float p0 = prow[ja], p1 = prow[ja + 1];
    float p2 = prow[jb], p3 = prow[jb + 1];
    float R0 = 0.f, R1 = 0.f, R2 = 0.f, R3 = 0.f;
#pragma unroll
    for (int k = 0; k < KB; ++k) {
      const float* vp = Vpad + (size_t)k * JF;
      float t = tk[k];
      R0 += t * vp[ja];
      R1 += t * vp[ja + 1];
      R2 += t * vp[jb];
      R3 += t * vp[jb + 1];
    }
    float i0 = fastrcp(R0), i1 = fastrcp(R1);
    float i2 = fastrcp(R2), i3 = fastrcp(R3);
    v2f an0 = { p0 * i0 * i0, p1 * i1 * i1 };
    v2f ad0 = { i0, i1 };
    v2f an1 = { p2 * i2 * i2, p3 * i3 * i3 };
    v2f ad1 = { i2, i3 };
    v2f bm0 = { brow[ja], brow[ja + 1] };
    v2f bm1 = { brow[jb], brow[jb + 1] };
    dnum0 = __builtin_amdgcn_wmma_f32_16x16x4_f32(false, an0, false, bm0,
                                                  (short)0, dnum0, false, false);
    dden0 = __builtin_amdgcn_wmma_f32_16x16x4_f32(false, ad0, false, bm0,
                                                  (short)0, dden0, false, false);
    dnum1 = __builtin_amdgcn_wmma_f32_16x16x4_f32(false, an1, false, bm1,
                                                  (short)0, dnum1, false, false);
    dden1 = __builtin_amdgcn_wmma_f32_16x16x4_f32(false, ad1, false, bm1,
                                                  (short)0, dden1, false, false);
  }
  v8f dnum = dnum0 + dnum1;
  v8f dden = dden0 + dden1;
  // D layout: lane holds column k=row, rows m = v + 8*hi (v = VGPR index)
  int k = row;
  if (k < KB) {
#pragma unroll
    for (int v = 0; v < 8; ++v) {
      int b2 = base + v + 8 * hi;
      if (b2 < IB) {
        size_t o = ((size_t)b2 * KB + k) * MCH + n;
        float tnew = T[o] * sqrtf(dnum[v] / dden[v]);
        T[o] = tnew > NMF_EPS ? tnew : NMF_EPS;
      }
    }
  }
}

__global__ void k_zero_nd(float* __restrict__ gnum, float* __restrict__ gden) {
  int idx = blockIdx.x * 256 + threadIdx.x;
  if (idx < KB * JF) { gnum[idx] = 0.f; gden[idx] = 0.f; }
}

// V-update accumulation: numV[k,j] = sum_i T[i,k]*(P/R^2)[i,j] (new T), split over I
__global__ void k_v_accum(const float* __restrict__ P, const float* __restrict__ T,
                          const float* __restrict__ Vpad, float* __restrict__ gnum,
                          float* __restrict__ gden, int n) {
  int j = blockIdx.x * 256 + threadIdx.x;
  if (j >= JF) return;
  const int CH = 129;                   // 16 chunks * 129 >= 2049
  int i0 = blockIdx.y * CH;
  int i1 = i0 + CH; if (i1 > IB) i1 = IB;
  float vk[KB], num[KB], den[KB];
#pragma unroll
  for (int k = 0; k < KB; ++k) {
    vk[k] = Vpad[(size_t)k * JF + j];
    num[k] = 0.f; den[k] = 0.f;
  }
  for (int i = i0; i < i1; ++i) {
    float tk[KB]; float R = 0.f;
#pragma unroll
    for (int k = 0; k < KB; ++k) {      // uniform addresses -> scalar loads
      tk[k] = T[((size_t)i * KB + k) * MCH + n];
      R += tk[k] * vk[k];
    }
    float invR = fastrcp(R);
    float pr2 = P[(size_t)i * JF + j] * invR * invR;
#pragma unroll
    for (int k = 0; k < KB; ++k) { num[k] += tk[k] * pr2; den[k] += tk[k] * invR; }
  }
#pragma unroll
  for (int k = 0; k < KB; ++k) {
    atomicAdd(&gnum[(size_t)k * JF + j], num[k]);
    atomicAdd(&gden[(size_t)k * JF + j], den[k]);
  }
}

__global__ void k_v_finalize(const float* __restrict__ gnum, const float* __restrict__ gden,
                             float* __restrict__ V, float* __restrict__ Vpad, int n) {
  int idx = blockIdx.x * 256 + threadIdx.x;
  if (idx >= KB * JF) return;
  int k = idx / JF, j = idx - k * JF;
  size_t ov = ((size_t)k * JF + j) * MCH + n;
  float vnew = V[ov] * sqrtf(gnum[idx] / gden[idx]);
  vnew = vnew > NMF_EPS ? vnew : NMF_EPS;
  V[ov] = vnew;
  Vpad[(size_t)k * JF + j] = vnew;      // keep padded slice in sync for k_w_update
}

// Per-bin weighted covariance D = (Xw Xw^H)/J, then A=W*Dreg, Cramer solve, W row n
__global__ void k_w_update(const float* __restrict__ Xre, const float* __restrict__ Xim,
                           const float* __restrict__ T, const float* __restrict__ Vpad,
                           float* __restrict__ W, int n) {
  __shared__ float red[4][256];
  int i = blockIdx.x;
  int t = threadIdx.x;
  float tk[KB];
#pragma unroll
  for (int k = 0; k < KB; ++k) tk[k] = T[((size_t)i * KB + k) * MCH + n];
  const float* x0r_p = Xre + ((size_t)i * 2 + 0) * JF;
  const float* x0i_p = Xim + ((size_t)i * 2 + 0) * JF;
  const float* x1r_p = Xre + ((size_t)i * 2 + 1) * JF;
  const float* x1i_p = Xim + ((size_t)i * 2 + 1) * JF;
  float d00 = 0.f, d11 = 0.f, d01r = 0.f, d01i = 0.f;
  for (int j = t; j < JF; j += 256) {
    float R = 0.f;
#pragma unroll
    for (int k = 0; k < KB; ++k) R += tk[k] * Vpad[(size_t)k * JF + j];
    float invR = fastrcp(R + IP_EPS);
    float a = x0r_p[j], b = x0i_p[j], c = x1r_p[j], d = x1i_p[j];
    d00  += (a * a + b * b) * invR;
    d11  += (c * c + d * d) * invR;
    d01r += (a * c + b * d) * invR;   // D01 = x0 * conj(x1)
    d01i += (b * c - a * d) * invR;
  }
  red[0][t] = d00; red[1][t] = d11; red[2][t] = d01r; red[3][t] = d01i;
  __syncthreads();
  for (int s = 128; s > 0; s >>= 1) {
    if (t < s) {
      red[0][t] += red[0][t + s];
      red[1][t] += red[1][t + s];
      red[2][t] += red[2][t + s];
      red[3][t] += red[3][t + s];
    }
    __syncthreads();
  }
  if (t == 0) {
    const float scale = 1.f / (float)JF;
    float D00  = red[0][0] * scale + IP_EPS;   // Dreg diagonal (real)
    float D11  = red[1][0] * scale + IP_EPS;
    float D01r = red[2][0] * scale;
    float D01i = red[3][0] * scale;            // D10 = conj(D01)
    float* wp = W + (size_t)i * 8;
    float A00r, A00i, A01r, A01i, A10r, A10i, A11r, A11i;
    {
      float w0r = wp[0], w0i = wp[1], w1r = wp[2], w1i = wp[3];
      A00r = w0r * D00 + w1r * D01r + w1i * D01i;
      A00i = w0i * D00 - w1r * D01i + w1i * D01r;
      A01r = w0r * D01r - w0i * D01i + w1r * D11;
      A01i = w0r * D01i + w0i * D01r + w1i * D11;
    }
    {
      float w0r = wp[4], w0i = wp[5], w1r = wp[6], w1i = wp[7];
      A10r = w0r * D00 + w1r * D01r + w1i * D01i;
      A10i = w0i * D00 - w1r * D01i + w1i * D01r;
      A11r = w0r * D01r - w0i * D01i + w1r * D11;
      A11i = w0r * D01i + w0i * D01r + w1i * D11;
    }
    float detr = A00r * A11r - A00i * A11i - (A01r * A10r - A01i * A10i);
    float deti = A00r * A11i + A00i * A11r - (A01r * A10i + A01i * A10r);
    float u0r, u0i, u1r, u1i;                  // column n of adj(A)
    if (n == 0) { u0r =  A11r; u0i =  A11i; u1r = -A10r; u1i = -A10i; }
    else        { u0r = -A01r; u0i = -A01i; u1r =  A00r; u1i =  A00i; }
    float invd2 = 1.f / (detr * detr + deti * deti);
    float b0r = (u0r * detr + u0i * deti) * invd2;
    float b0i = (u0i * detr - u0r * deti) * invd2;
    float b1r = (u1r * detr + u1i * deti) * invd2;
    float b1i = (u1i * detr - u1r * deti) * invd2;
    float zr = D01r * b1r - D01i * b1i;        // D01 * b1
    float zi = D01r * b1i + D01i * b1r;
    float quad = (b0r * b0r + b0i * b0i) * D00 + (b1r * b1r + b1i * b1i) * D11
               + 2.f * (b0r * zr + b0i * zi);
    float inv = 1.f / sqrtf(quad + IP_EPS);
    wp[(n * 2 + 0) * 2]     =  b0r * inv;      // W[i,n,:] = conj(b)/denom
    wp[(n * 2 + 0) * 2 + 1] = -b0i * inv;
    wp[(n * 2 + 1) * 2]     =  b1r * inv;
    wp[(n * 2 + 1) * 2 + 1] = -b1i * inv;
  }
}

// out (N,J,I,2) = W @ Xc, LDS-tiled transpose so loads (j) and stores (i) coalesce
__global__ void k_final_out(const float* __restrict__ Xre, const float* __restrict__ Xim,
                            const float* __restrict__ W, float* __restrict__ out) {
  __shared__ float syr[32][33];
  __shared__ float syi[32][33];
  int n  = blockIdx.z;
  int i0 = blockIdx.y * 32;
  int j0 = blockIdx.x * 32;
  int tx = threadIdx.x, ty = threadIdx.y;
  for (int s = 0; s < 4; ++s) {
    int il = ty + 8 * s;
    int i = i0 + il; int ic = i < IB ? i : IB - 1;
    int j = j0 + tx; int jc = j < JF ? j : JF - 1;
    const float* w = W + (size_t)ic * 8 + (size_t)n * 4;
    float w0r = w[0], w0i = w[1], w1r = w[2], w1i = w[3];
    size_t o0 = ((size_t)ic * 2 + 0) * JF + jc;
    size_t o1 = ((size_t)ic * 2 + 1) * JF + jc;
    float a = Xre[o0], b = Xim[o0], c = Xre[o1], d = Xim[o1];
    syr[il][tx] = w0r * a - w0i * b + w1r * c - w1i * d;
    syi[il][tx] = w0r * b + w0i * a + w1r * d + w1i * c;
  }
  __syncthreads();
  for (int s = 0; s < 4; ++s) {
    int jl = ty + 8 * s;
    int i = i0 + tx, j = j0 + jl;
    if (i < IB && j < JF) {
      size_t o = (((size_t)n * JF + j) * IB + i) * 2;
      out[o]     = syr[tx][jl];
      out[o + 1] = syi[tx][jl];
    }
  }
}

// ---------------------------------------------------------------------------
extern "C" void kernel_launch(void* const* d_in, const int* in_sizes, int n_in,
                              void* d_out, int out_size, void* d_ws, size_t ws_size,
                              hipStream_t stream) {
  const float* X  = (const float*)d_in[0];   // (M,J,I,2)
  const float* T0 = (const float*)d_in[1];   // (I,K,M)
  const float* V0 = (const float*)d_in[2];   // (K,J,M)
  float* out = (float*)d_out;                // (N,J,I,2)
  float* ws = (float*)d_ws;

  size_t off = 0;
  float* Xre  = ws + off; off += (size_t)IB * MCH * JF;   // 8.196M
  float* Xim  = ws + off; off += (size_t)IB * MCH * JF;   // 8.196M
  float* P    = ws + off; off += (size_t)IB * JF;         // 4.098M
  float* W    = ws + off; off += (size_t)IB * 8;
  float* T    = ws + off; off += (size_t)IB * KB * MCH;
  float* V    = ws + off; off += (size_t)KB * JF * MCH;
  float* Vpad = ws + off; off += (size_t)16 * JF;
  float* gnum = ws + off; off += (size_t)KB * JF;
  float* gden = ws + off; off += (size_t)KB * JF;         // total ~82.5 MB

  dim3 tb(32, 8, 1);
  dim3 tg((JF + 31) / 32, (IB + 31) / 32, MCH);
  k_prep_x<<<tg, tb, 0, stream>>>(X, Xre, Xim);
  k_init_w<<<(IB + 255) / 256, 256, 0, stream>>>(W);
  hipMemcpyAsync(T, T0, sizeof(float) * (size_t)IB * KB * MCH,
                 hipMemcpyDeviceToDevice, stream);
  hipMemcpyAsync(V, V0, sizeof(float) * (size_t)KB * JF * MCH,
                 hipMemcpyDeviceToDevice, stream);

  for (int it = 0; it < NITER; ++it) {
    for (int n = 0; n < MCH; ++n) {
      k_compute_p<<<dim3((JF + 255) / 256, IB, 1), 256, 0, stream>>>(Xre, Xim, W, P, n);
      k_pack_vpad<<<(16 * JF + 255) / 256, 256, 0, stream>>>(V, Vpad, n);
      k_t_update<<<(IB + 63) / 64, 128, 0, stream>>>(P, Vpad, T, n);
      k_zero_nd<<<(KB * JF + 255) / 256, 256, 0, stream>>>(gnum, gden);
      k_v_accum<<<dim3((JF + 255) / 256, 16, 1), 256, 0, stream>>>(P, T, Vpad, gnum, gden, n);
      k_v_finalize<<<(KB * JF + 255) / 256, 256, 0, stream>>>(gnum, gden, V, Vpad, n);
      k_w_update<<<IB, 256, 0, stream>>>(Xre, Xim, T, Vpad, W, n);
    }
  }
  k_final_out<<<tg, tb, 0, stream>>>(Xre, Xim, W, out);
}